// MQCCLayer_29145648071140
// MI455X (gfx1250) — compile-verified
//
#include <hip/hip_runtime.h>
#include <hip/hip_bf16.h>
#include <math.h>

// Problem constants (from reference)
#define CIN    4
#define COUT   8
#define NF     2
#define HH     512
#define WW     512

// Tiling
#define TILE_X 128            // pixels per block along x (8 waves * 16)
#define ROWS   8              // output rows per block
#define LROWS  (ROWS + 2)     // staged input rows (with halo) = 10
#define PITCH  136            // padded LDS row pitch (floats); interior at col 4 (16B aligned)
// LDS cols: [3] = left halo (gx = x0-1), [4..131] = interior, [132] = right halo

typedef __attribute__((ext_vector_type(2))) float v2f;
typedef __attribute__((ext_vector_type(4))) float v4f;
typedef __attribute__((ext_vector_type(8))) float v8f;

// ---------------------------------------------------------------------------
// Tiny pre-kernel: wn[f][t] = angles[f][t] / ||angles[f]||_2   (18 floats)
// ---------------------------------------------------------------------------
__global__ void mqcc_normalize_angles(const float* __restrict__ angles,
                                      float* __restrict__ wn) {
    if (threadIdx.x == 0 && blockIdx.x == 0) {
        #pragma unroll
        for (int f = 0; f < NF; ++f) {
            float ss = 0.0f;
            #pragma unroll
            for (int i = 0; i < 9; ++i) {
                float v = angles[f * 9 + i];
                ss += v * v;
            }
            float inv = 1.0f / sqrtf(ss);
            #pragma unroll
            for (int i = 0; i < 9; ++i)
                wn[f * 9 + i] = angles[f * 9 + i] * inv;
        }
    }
}

// ---------------------------------------------------------------------------
// Main kernel: depthwise 3x3 conv (4 in-ch -> 8 out-ch, 2 shared filters)
// as a block-diagonal GEMM on V_WMMA_F32_16X16X4_F32:
//   D[pixel(16), ch(16)] += A[pixel(16), k(4)] x B[k(4), ch(16)]
// K = 36 taps (c,kh,kw) -> 9 WMMA steps of K=4 per 16-pixel strip per row.
// ---------------------------------------------------------------------------
__global__ __launch_bounds__(256) void mqcc_conv_wmma(
        const float* __restrict__ x,    // (B, 4, 512, 512)
        const float* __restrict__ wn,   // (2, 3, 3) normalized
        float* __restrict__ out) {      // (B, 8, 512, 512)

    __shared__ float lds[CIN * LROWS * PITCH];

    const int tid = threadIdx.x;
    const int x0  = blockIdx.x * TILE_X;   // 0..384 (multiple of 128)
    const int y0  = blockIdx.y * ROWS;     // 0..504 (multiple of 8)
    const int b   = blockIdx.z;

    const int lane = tid & 31;
    const int wave = tid >> 5;             // 0..7

    // ---- stage input tile (4 ch, 10 rows, 130 cols) into LDS, zero halo ----
    const float* xb = x + (size_t)b * CIN * HH * WW;

    // interior: one wave per staged row, lane-indexed float4 (global_load_b128)
    for (int row = wave; row < CIN * LROWS; row += 8) {
        int c  = row / LROWS;
        int r  = row - c * LROWS;
        int gy = y0 - 1 + r;
        v4f v = {0.f, 0.f, 0.f, 0.f};
        if (gy >= 0 && gy < HH)
            v = *(const v4f*)(xb + ((size_t)c * HH + gy) * WW + x0 + lane * 4);
        *(v4f*)&lds[row * PITCH + 4 + lane * 4] = v;
    }
    // halos: 2 scalars per staged row (80 threads)
    if (tid < CIN * LROWS * 2) {
        int row  = tid >> 1;
        int side = tid & 1;
        int c  = row / LROWS;
        int r  = row - c * LROWS;
        int gy = y0 - 1 + r;
        int gx = side ? (x0 + TILE_X) : (x0 - 1);
        float v = 0.f;
        if (gy >= 0 && gy < HH && gx >= 0 && gx < WW)
            v = xb[((size_t)c * HH + gy) * WW + gx];
        lds[row * PITCH + (side ? (4 + TILE_X) : 3)] = v;
    }
    __syncthreads();

    // ---- per-lane setup ----
    const int m     = lane & 15;       // A row = pixel-in-strip; B col = out channel
    const int hi    = lane >> 4;       // half-wave select
    const int xbase = wave * 16;       // 16-pixel strip per wave

    // ISA 16x16x4 f32 layout: this lane's A/B VGPR0 holds K = 4s + 2*hi,
    // VGPR1 holds K+1.  Wfull[n, k=c*9+kh*3+kw] nonzero iff n<8 && c==n/2.
    // LDS tap col for output x = x0+xbase+m, tap kw: (xbase+m+kw-1)+4.
    float wb0[9], wb1[9];
    int   ba0[9], ba1[9];
    #pragma unroll
    for (int s = 0; s < 9; ++s) {
        int k0 = 4 * s + 2 * hi;
        int k1 = k0 + 1;
        {
            int c = k0 / 9, t = k0 - c * 9;
            int kh = t / 3, kw = t - kh * 3;
            wb0[s] = (m < COUT && c == (m >> 1)) ? wn[(m & 1) * 9 + t] : 0.0f;
            ba0[s] = (c * LROWS + kh) * PITCH + xbase + m + kw + 3;
        }
        {
            int c = k1 / 9, t = k1 - c * 9;
            int kh = t / 3, kw = t - kh * 3;
            wb1[s] = (m < COUT && c == (m >> 1)) ? wn[(m & 1) * 9 + t] : 0.0f;
            ba1[s] = (c * LROWS + kh) * PITCH + xbase + m + kw + 3;
        }
    }

    float* outb = out + (size_t)b * COUT * HH * WW;

    #pragma unroll
    for (int rr = 0; rr < ROWS; ++rr) {
        const int roff = rr * PITCH;
        v8f acc = {0.f, 0.f, 0.f, 0.f, 0.f, 0.f, 0.f, 0.f};

        #pragma unroll
        for (int s = 0; s < 9; ++s) {
            v2f a, bm;
            a.x  = lds[ba0[s] + roff];   // A[m, 4s + 2*hi]
            a.y  = lds[ba1[s] + roff];   // A[m, 4s + 2*hi + 1]
            bm.x = wb0[s];               // B[4s + 2*hi, n]
            bm.y = wb1[s];               // B[4s + 2*hi + 1, n]
            acc = __builtin_amdgcn_wmma_f32_16x16x4_f32(
                    /*neg_a=*/false, a, /*neg_b=*/false, bm,
                    /*c_mod=*/(short)0, acc,
                    /*reuse_a=*/false, /*reuse_b=*/false);
        }

        // D: lane n (=m) holds channel n, pixels M = v + 8*hi (v = 0..7)
        if (m < COUT) {
            const int y = y0 + rr;
            float* p = outb + ((size_t)m * HH + y) * WW + x0 + xbase + hi * 8;
            v4f lo4 = {acc[0], acc[1], acc[2], acc[3]};
            v4f hi4 = {acc[4], acc[5], acc[6], acc[7]};
            *(v4f*)(p)     = lo4;   // offsets are multiples of 8 floats -> 16B aligned
            *(v4f*)(p + 4) = hi4;
        }
    }
}

// ---------------------------------------------------------------------------
extern "C" void kernel_launch(void* const* d_in, const int* in_sizes, int n_in,
                              void* d_out, int out_size, void* d_ws, size_t ws_size,
                              hipStream_t stream) {
    const float* x      = (const float*)d_in[0];   // (B, 4, 512, 512) f32
    const float* angles = (const float*)d_in[1];   // (2, 3, 3) f32
    float* out = (float*)d_out;                    // (B, 8, 512, 512) f32
    float* wn  = (float*)d_ws;                     // 18 floats scratch

    const int nb = in_sizes[0] / (CIN * HH * WW);  // batch (32)

    mqcc_normalize_angles<<<1, 32, 0, stream>>>(angles, wn);

    dim3 grid(WW / TILE_X, HH / ROWS, nb);         // (4, 64, 32)
    mqcc_conv_wmma<<<grid, 256, 0, stream>>>(x, wn, out);
}